// Encoder_48567490183709
// MI455X (gfx1250) — compile-verified
//
#include <hip/hip_runtime.h>

// fp32 WMMA fragment types (wave32): A/B 16x4 & 4x16 = 2 VGPRs, C/D 16x16 = 8 VGPRs
typedef __attribute__((ext_vector_type(2))) float v2f;
typedef __attribute__((ext_vector_type(8))) float v8f;

// ---------------- degree / normalization ----------------

__global__ __launch_bounds__(256) void k_init_deg(float* __restrict__ deg, int n) {
    int i = blockIdx.x * blockDim.x + threadIdx.x;
    if (i < n) deg[i] = 1.0f;  // self-loop contributes 1 to every node's degree
}

__global__ __launch_bounds__(256) void k_edge_deg(const int* __restrict__ dst,
                                                  float* __restrict__ deg, int E) {
    int e = blockIdx.x * blockDim.x + threadIdx.x;
    if (e < E) atomicAdd(&deg[dst[e]], 1.0f);
}

__global__ __launch_bounds__(256) void k_deg_to_dinv(float* __restrict__ deg, int n) {
    int i = blockIdx.x * blockDim.x + threadIdx.x;
    if (i < n) {
        float d = deg[i];
        deg[i] = (d > 0.0f) ? rsqrtf(d) : 0.0f;
    }
}

// ---------------- dense GEMM via V_WMMA_F32_16X16X4_F32 ----------------
// One wave computes a 16 x (NT*16) tile of Out = A[M,K] @ W[K,N], N = NT*16.
// Optional ReLU applied to A-operand loads (fuses relu(agg1) for layer 2).

template <int NT, bool RELU>
__global__ __launch_bounds__(256) void k_gemm_wmma(const float* __restrict__ A,
                                                   const float* __restrict__ W,
                                                   float* __restrict__ Out,
                                                   int M, int K) {
    constexpr int N = NT * 16;
    const int wave = blockIdx.x * (blockDim.x >> 5) + (threadIdx.x >> 5);
    const int lane = threadIdx.x & 31;
    const int numTiles = (M + 15) >> 4;
    if (wave >= numTiles) return;  // wave-uniform: EXEC stays all-1s for WMMA

    const int r    = wave << 4;
    const int m    = lane & 15;          // row within A tile / column within B tile
    const int koff = (lane >> 4) << 1;   // lanes 16-31 hold K+2, K+3 slices
    const int arow = (r + m < M) ? (r + m) : (M - 1);

    v8f zero = {};
    v8f acc[NT];
#pragma unroll
    for (int t = 0; t < NT; ++t) acc[t] = zero;

    for (int k = 0; k < K; k += 4) {
        // A 16x4 fragment: v0 = K=koff, v1 = K=koff+1 (per ISA 32-bit A layout)
        const float* ap = A + (size_t)arow * K + (k + koff);
        v2f a;
        a.x = ap[0];
        a.y = ap[1];
        if (RELU) {
            a.x = fmaxf(a.x, 0.0f);
            a.y = fmaxf(a.y, 0.0f);
        }
#pragma unroll
        for (int t = 0; t < NT; ++t) {
            // B 4x16 fragment: row K striped across lanes, K/K+half split at lane 16
            const float* bp = W + (size_t)(k + koff) * N + t * 16 + m;
            v2f b;
            b.x = bp[0];
            b.y = bp[N];
            acc[t] = __builtin_amdgcn_wmma_f32_16x16x4_f32(
                false, a, false, b, (short)0, acc[t], false, false);
        }
    }

    // C/D layout: VGPR g -> row r+g (lanes 0-15) / r+g+8 (lanes 16-31), col = lane&15
    const int half = (lane >> 4) << 3;
#pragma unroll
    for (int t = 0; t < NT; ++t) {
#pragma unroll
        for (int g = 0; g < 8; ++g) {
            int row = r + half + g;
            if (row < M) Out[(size_t)row * N + t * 16 + m] = acc[t][g];
        }
    }
}

// ---------------- aggregation (scatter-add) ----------------
// Agg[i,:] = bias + dinv[i]^2 * H[i,:]  (self-loop term + bias folded into init)

template <int N>
__global__ __launch_bounds__(256) void k_init_agg(const float* __restrict__ H,
                                                  const float* __restrict__ bias,
                                                  const float* __restrict__ dinv,
                                                  float* __restrict__ Agg, int n) {
    long long idx = (long long)blockIdx.x * blockDim.x + threadIdx.x;
    if (idx >= (long long)n * N) return;
    int node = (int)(idx / N);
    int c    = (int)(idx % N);
    float di = dinv[node];
    Agg[idx] = bias[c] + di * di * H[idx];
}

// One wave per edge: lanes gather a contiguous float4/float2 slice of H[src,:]
// (coalesced 512B / 256B per edge) and scatter with global fp32 atomics.
template <int N>
__global__ __launch_bounds__(256) void k_edge_agg(const int* __restrict__ src,
                                                  const int* __restrict__ dst,
                                                  const float* __restrict__ H,
                                                  const float* __restrict__ dinv,
                                                  float* __restrict__ Agg, int E) {
    const int wave = blockIdx.x * (blockDim.x >> 5) + (threadIdx.x >> 5);
    const int lane = threadIdx.x & 31;
    if (wave >= E) return;
    const int s = src[wave];
    const int d = dst[wave];
    const float norm = dinv[s] * dinv[d];

    constexpr int V = N / 32;
    const float* hp = H + (size_t)s * N + lane * V;
    float* op = Agg + (size_t)d * N + lane * V;

    float vals[V];
    if constexpr (V == 4) {
        float4 t = *reinterpret_cast<const float4*>(hp);
        vals[0] = t.x; vals[1] = t.y; vals[2] = t.z; vals[3] = t.w;
    } else {
        float2 t = *reinterpret_cast<const float2*>(hp);
        vals[0] = t.x; vals[1] = t.y;
    }
#pragma unroll
    for (int i = 0; i < V; ++i) atomicAdd(op + i, vals[i] * norm);
}

// ---------------- driver ----------------

extern "C" void kernel_launch(void* const* d_in, const int* in_sizes, int n_in,
                              void* d_out, int out_size, void* d_ws, size_t ws_size,
                              hipStream_t stream) {
    (void)n_in; (void)out_size; (void)ws_size;

    const float* x  = (const float*)d_in[0];
    const int*   ei = (const int*)d_in[1];   // [2, E] int32 (jax x64 disabled)
    const float* W1 = (const float*)d_in[2];
    const float* b1 = (const float*)d_in[3];
    const float* W2 = (const float*)d_in[4];
    const float* b2 = (const float*)d_in[5];

    const int C_IN = 128, C_HID = 128, C_OUT = 64;
    const int n = in_sizes[0] / C_IN;   // 50000
    const int E = in_sizes[1] / 2;      // 800000
    const int* srcv = ei;
    const int* dstv = ei + E;

    // workspace carve-out (256B aligned): dinv | h | agg1 | g  (~64 MB total)
    char* ws = (char*)d_ws;
    size_t o0 = 0;
    size_t o1 = o0 + (((size_t)n * 4 + 255) & ~(size_t)255);
    size_t o2 = o1 + (((size_t)n * C_HID * 4 + 255) & ~(size_t)255);
    size_t o3 = o2 + (((size_t)n * C_HID * 4 + 255) & ~(size_t)255);
    float* dinv = (float*)(ws + o0);
    float* h    = (float*)(ws + o1);
    float* agg1 = (float*)(ws + o2);
    float* g    = (float*)(ws + o3);
    float* out  = (float*)d_out;

    const int numTiles   = (n + 15) / 16;
    const int gemmBlocks = (numTiles + 7) / 8;          // 8 waves / 256-thread block
    const int edgeBlocks = (E + 7) / 8;                 // 1 wave per edge

    // degree + symmetric normalization (shared by both layers)
    k_init_deg<<<(n + 255) / 256, 256, 0, stream>>>(dinv, n);
    k_edge_deg<<<(E + 255) / 256, 256, 0, stream>>>(dstv, dinv, E);
    k_deg_to_dinv<<<(n + 255) / 256, 256, 0, stream>>>(dinv, n);

    // layer 1: h = x @ W1 ; agg1 = b1 + D^-1/2 A D^-1/2 h
    k_gemm_wmma<8, false><<<gemmBlocks, 256, 0, stream>>>(x, W1, h, n, C_IN);
    k_init_agg<128><<<(unsigned)(((long long)n * 128 + 255) / 256), 256, 0, stream>>>(
        h, b1, dinv, agg1, n);
    k_edge_agg<128><<<edgeBlocks, 256, 0, stream>>>(srcv, dstv, h, dinv, agg1, E);

    // layer 2: g = relu(agg1) @ W2 ; out = b2 + D^-1/2 A D^-1/2 g
    k_gemm_wmma<4, true><<<gemmBlocks, 256, 0, stream>>>(agg1, W2, g, n, C_HID);
    k_init_agg<64><<<(unsigned)(((long long)n * 64 + 255) / 256), 256, 0, stream>>>(
        g, b2, dinv, out, n);
    k_edge_agg<64><<<edgeBlocks, 256, 0, stream>>>(srcv, dstv, g, dinv, out, E);
}